// VariationalGCNEncoder_57071525429451
// MI455X (gfx1250) — compile-verified
//
#include <hip/hip_runtime.h>

typedef __attribute__((ext_vector_type(2))) float v2f;
typedef __attribute__((ext_vector_type(8))) float v8f;

#define N_NODES 50000
#define N_EDGES 800000
#define F_IN    128
#define F_HID   128
#define F_OUT   64

// ---------------------------------------------------------------- utilities

__global__ void zero_f32_kernel(float* __restrict__ p, int n) {
  int i = blockIdx.x * blockDim.x + threadIdx.x;
  if (i < n) p[i] = 0.0f;
}

// deg[dst] += 1 over real edges (self-loops folded in later as +1)
__global__ void degree_kernel(const int* __restrict__ dst, float* __restrict__ deg) {
  int e = blockIdx.x * blockDim.x + threadIdx.x;
  if (e < N_EDGES) unsafeAtomicAdd(&deg[dst[e]], 1.0f);
}

// in-place: deg -> dinv = rsqrt(deg + 1)   (+1 = self loop, guarantees > 0)
__global__ void dinv_kernel(float* __restrict__ deg) {
  int i = blockIdx.x * blockDim.x + threadIdx.x;
  if (i < N_NODES) deg[i] = rsqrtf(deg[i] + 1.0f);
}

// agg[i,f] = feat[i,f] * dinv[i]^2  (self-loop contribution; also zero-inits agg)
__global__ void selfloop_init_kernel(const float* __restrict__ feat,
                                     const float* __restrict__ dinv,
                                     float* __restrict__ agg) {
  int i = blockIdx.x * blockDim.x + threadIdx.x;   // over N*128
  if (i < N_NODES * F_HID) {
    float di = dinv[i >> 7];                       // /128
    agg[i] = feat[i] * di * di;
  }
}

// agg[dst] += feat[src] * dinv[src]*dinv[dst]; one wave (32 lanes) per edge,
// 4 contiguous floats per lane (float4 gather, 4 f32 atomics -> L2 resident)
__global__ void edge_agg_kernel(const int* __restrict__ src,
                                const int* __restrict__ dst,
                                const float* __restrict__ dinv,
                                const float* __restrict__ feat,
                                float* __restrict__ agg) {
  int t  = blockIdx.x * blockDim.x + threadIdx.x;
  int e  = t >> 5;                  // edge index
  int fq = (t & 31) << 2;           // feature offset (0..124 step 4)
  if (e >= N_EDGES) return;
  int s = src[e];
  int d = dst[e];
  float nrm = dinv[s] * dinv[d];
  const float4 v = *(const float4*)(feat + (size_t)s * F_HID + fq);
  float* o = agg + (size_t)d * F_HID + fq;
  unsafeAtomicAdd(o + 0, v.x * nrm);
  unsafeAtomicAdd(o + 1, v.y * nrm);
  unsafeAtomicAdd(o + 2, v.z * nrm);
  unsafeAtomicAdd(o + 3, v.w * nrm);
}

// ---------------------------------------------------------------- WMMA GEMMs
// A-frag (16x4 f32): lane = 16*kh + m, vgpr v holds A[m][kh*2+v]
// B-frag (4x16 f32): lane = 16*kh + n, vgpr v holds B[kh*2+v][n]
// C/D (16x16 f32):   vgpr j, lanes 0-15 -> M=j, lanes 16-31 -> M=8+j; N=lane&15

// H = relu(A[50000,128] @ W[128,128] + bias)
__global__ void gemm_relu_kernel(const float* __restrict__ A,
                                 const float* __restrict__ W,
                                 const float* __restrict__ bias,
                                 float* __restrict__ H) {
  __shared__ float As[16 * 128];                  // 8 KB A-tile shared by 8 waves
  const int m0 = blockIdx.x << 4;                 // 3125 blocks * 16 rows = 50000
  for (int i = threadIdx.x; i < 16 * 128; i += 256)
    As[i] = A[(size_t)m0 * 128 + i];
  __syncthreads();

  const int lane = threadIdx.x & 31;
  const int wave = threadIdx.x >> 5;              // 8 waves -> 8 N-tiles (N=128)
  const int n0   = wave << 4;
  const int am   = lane & 15;
  const int kh   = lane >> 4;

  const float* arow = As + am * 128 + (kh << 1);
  const float* bptr = W + (kh << 1) * 128 + n0 + am;

  v8f acc = {};
#pragma unroll
  for (int k0 = 0; k0 < 128; k0 += 4) {
    v2f a; a.x = arow[0];   a.y = arow[1];
    v2f b; b.x = bptr[0];   b.y = bptr[128];
    acc = __builtin_amdgcn_wmma_f32_16x16x4_f32(false, a, false, b,
                                                (short)0, acc, false, false);
    arow += 4;
    bptr += 4 * 128;
  }

  const int col   = n0 + am;
  const float bv  = bias[col];
  const int rbase = m0 + (kh << 3);
#pragma unroll
  for (int j = 0; j < 8; ++j) {
    float v = acc[j] + bv;
    H[(size_t)(rbase + j) * 128 + col] = v > 0.0f ? v : 0.0f;
  }
}

// mu = A @ Wmu + bmu ; ls = A @ Wls + bls   (A:[50000,128], W:[128,64])
// waves 0-3 compute mu N-tiles, waves 4-7 compute logstd N-tiles
__global__ void gemm_dual_kernel(const float* __restrict__ A,
                                 const float* __restrict__ Wmu,
                                 const float* __restrict__ bmu,
                                 const float* __restrict__ Wls,
                                 const float* __restrict__ bls,
                                 float* __restrict__ mu,
                                 float* __restrict__ ls) {
  __shared__ float As[16 * 128];
  const int m0 = blockIdx.x << 4;
  for (int i = threadIdx.x; i < 16 * 128; i += 256)
    As[i] = A[(size_t)m0 * 128 + i];
  __syncthreads();

  const int lane  = threadIdx.x & 31;
  const int wave  = threadIdx.x >> 5;
  const bool isls = wave >= 4;                    // wave-uniform: EXEC stays all-1
  const float* W    = isls ? Wls : Wmu;
  const float* bias = isls ? bls : bmu;
  float*       out  = isls ? ls  : mu;
  const int n0 = (wave & 3) << 4;                 // 4 tiles * 16 = 64 cols
  const int am = lane & 15;
  const int kh = lane >> 4;

  const float* arow = As + am * 128 + (kh << 1);
  const float* bptr = W + (kh << 1) * 64 + n0 + am;

  v8f acc = {};
#pragma unroll
  for (int k0 = 0; k0 < 128; k0 += 4) {
    v2f a; a.x = arow[0]; a.y = arow[1];
    v2f b; b.x = bptr[0]; b.y = bptr[64];
    acc = __builtin_amdgcn_wmma_f32_16x16x4_f32(false, a, false, b,
                                                (short)0, acc, false, false);
    arow += 4;
    bptr += 4 * 64;
  }

  const int col   = n0 + am;
  const float bv  = bias[col];
  const int rbase = m0 + (kh << 3);
#pragma unroll
  for (int j = 0; j < 8; ++j)
    out[(size_t)(rbase + j) * 64 + col] = acc[j] + bv;
}

// ---------------------------------------------------------------- launcher

extern "C" void kernel_launch(void* const* d_in, const int* in_sizes, int n_in,
                              void* d_out, int out_size, void* d_ws, size_t ws_size,
                              hipStream_t stream) {
  const float* x   = (const float*)d_in[0];
  const int*   ei  = (const int*)d_in[1];      // [2, E]: row0 = src, row1 = dst
  const float* W1  = (const float*)d_in[2];
  const float* b1  = (const float*)d_in[3];
  const float* Wmu = (const float*)d_in[4];
  const float* bmu = (const float*)d_in[5];
  const float* Wls = (const float*)d_in[6];
  const float* bls = (const float*)d_in[7];

  const int* src = ei;
  const int* dst = ei + N_EDGES;

  float* mu = (float*)d_out;
  float* ls = mu + (size_t)N_NODES * F_OUT;

  // workspace: deg/dinv (50000 f32, padded), agg (50000x128), h (50000x128)
  float* deg = (float*)d_ws;
  float* agg = deg + 65536;
  float* h   = agg + (size_t)N_NODES * F_HID;

  const int T = 256;

  // degree -> dinv
  zero_f32_kernel<<<(N_NODES + T - 1) / T, T, 0, stream>>>(deg, N_NODES);
  degree_kernel<<<(N_EDGES + T - 1) / T, T, 0, stream>>>(dst, deg);
  dinv_kernel<<<(N_NODES + T - 1) / T, T, 0, stream>>>(deg);

  // layer 1: aggregate x, then h = relu(agg @ W1 + b1)
  selfloop_init_kernel<<<(N_NODES * F_IN + T - 1) / T, T, 0, stream>>>(x, deg, agg);
  edge_agg_kernel<<<(N_EDGES * 32) / T, T, 0, stream>>>(src, dst, deg, x, agg);
  gemm_relu_kernel<<<N_NODES / 16, T, 0, stream>>>(agg, W1, b1, h);

  // layer 2: aggregate h once, then project to mu and logstd
  selfloop_init_kernel<<<(N_NODES * F_HID + T - 1) / T, T, 0, stream>>>(h, deg, agg);
  edge_agg_kernel<<<(N_EDGES * 32) / T, T, 0, stream>>>(src, dst, deg, h, agg);
  gemm_dual_kernel<<<N_NODES / 16, T, 0, stream>>>(agg, Wmu, bmu, Wls, bls, mu, ls);
}